// ActorCriticLSTM_43112881717795
// MI455X (gfx1250) — compile-verified
//
#include <hip/hip_runtime.h>
#include <math.h>
#include <stdint.h>

#define TT 256
#define BB 1024
#define OO 128
#define HH 128
#define AAD 8
#define G4 512
#define LDP 132            // padded LDS row pitch (floats): 16B-aligned rows, conflict-free b64 reads

#define LOG2PI_F 1.8378770664093453f

#define AS1 __attribute__((address_space(1)))
#define AS3 __attribute__((address_space(3)))

typedef __attribute__((ext_vector_type(2))) float v2f;
typedef __attribute__((ext_vector_type(8))) float v8f;
typedef int v4i_ __attribute__((vector_size(16)));   // matches builtin's b128 pointee type

__device__ __forceinline__ v8f wmma4(v2f a, v2f b, v8f c) {
  // D = A(16x4,f32) x B(4x16,f32) + C(16x16,f32)
  return __builtin_amdgcn_wmma_f32_16x16x4_f32(false, a, false, b, (short)0, c, false, false);
}

__device__ __forceinline__ float fast_tanh(float x) {
#if __has_builtin(__builtin_amdgcn_tanhf)
  return __builtin_amdgcn_tanhf(x);         // v_tanh_f32
#elif __has_builtin(__builtin_amdgcn_tanh_f32)
  return __builtin_amdgcn_tanh_f32(x);
#else
  return tanhf(x);
#endif
}

__device__ __forceinline__ float sigm(float x) {
  return __builtin_amdgcn_rcpf(1.f + __expf(-x));   // v_exp_f32 + v_rcp_f32, branch-free
}

// ---- CDNA5 async global<->LDS copies (ASYNCcnt) ----
__device__ __forceinline__ void g2l_async_b128(const float* g, float* l) {
#if __has_builtin(__builtin_amdgcn_global_load_async_to_lds_b128)
  __builtin_amdgcn_global_load_async_to_lds_b128(
      (AS1 v4i_*)(uintptr_t)g, (AS3 v4i_*)l, 0, 0);
#else
  unsigned loff = (unsigned)(uintptr_t)(AS3 float*)l;
  asm volatile("global_load_async_to_lds_b128 %0, %1, off"
               :: "v"(loff), "v"((unsigned long long)(uintptr_t)g) : "memory");
#endif
}

__device__ __forceinline__ void l2g_async_b128(float* g, const float* l) {
#if __has_builtin(__builtin_amdgcn_global_store_async_from_lds_b128)
  __builtin_amdgcn_global_store_async_from_lds_b128(
      (AS1 v4i_*)(uintptr_t)g, (AS3 v4i_*)(uintptr_t)l, 0, 0);
#else
  unsigned loff = (unsigned)(uintptr_t)(AS3 const float*)l;
  asm volatile("global_store_async_from_lds_b128 %0, %1, off"
               :: "v"((unsigned long long)(uintptr_t)g), "v"(loff) : "memory");
#endif
}

__device__ __forceinline__ void wait_async0() {
#if __has_builtin(__builtin_amdgcn_s_wait_asynccnt)
  __builtin_amdgcn_s_wait_asynccnt(0);
#else
  asm volatile("s_wait_asynccnt 0x0" ::: "memory");
#endif
}

// One 16-row strip per wave: OUT[strip][j0..j0+16*nt_count) = act( IN[strip] @ W^T + bias )
// Processes two 16-col output tiles per pass (A operand reuse, 2 independent WMMA chains).
__device__ __forceinline__ void layer_pass(
    const float* __restrict__ sIn, const float* __restrict__ W,
    const float* __restrict__ bias, float* __restrict__ sOut,
    int wave, int lane, int jbase, int nt_count, bool do_tanh)
{
  const int lh = lane & 15, hl = lane >> 4;
  const float* Ab = sIn + (wave * 16 + lh) * LDP + (hl << 1);
  for (int nt = 0; nt < nt_count; nt += 2) {
    const int j0 = jbase + nt * 16;
    const float* B0 = W + (size_t)(j0 + lh) * 128 + (hl << 1);
    const float* B1 = B0 + 16 * 128;
    v8f acc0 = {}, acc1 = {};
#pragma unroll 4
    for (int kb = 0; kb < 128; kb += 4) {
      v2f a = *(const v2f*)(Ab + kb);
      acc0 = wmma4(a, *(const v2f*)(B0 + kb), acc0);
      acc1 = wmma4(a, *(const v2f*)(B1 + kb), acc1);
    }
    float bv0 = bias[j0 + lh], bv1 = bias[j0 + 16 + lh];
#pragma unroll
    for (int v = 0; v < 8; ++v) {
      float r0v = acc0[v] + bv0;
      float r1v = acc1[v] + bv1;
      if (do_tanh) { r0v = fast_tanh(r0v); r1v = fast_tanh(r1v); }
      float* orow = sOut + (wave * 16 + v + 8 * hl) * LDP + lh;
      orow[nt * 16]        = r0v;
      orow[nt * 16 + 16]   = r1v;
    }
  }
}

// ---------------- Encoder + input-gate precompute: xg[T*B,512] ----------------
__global__ void __launch_bounds__(256) enc_xg_kernel(
    const float* __restrict__ obs,
    const float* __restrict__ w1, const float* __restrict__ b1,
    const float* __restrict__ w2, const float* __restrict__ b2,
    const float* __restrict__ wih, const float* __restrict__ bih,
    float* __restrict__ xg)
{
  __shared__ float sA[128 * LDP];
  __shared__ float sB[128 * LDP];
  const int tid = threadIdx.x, lane = tid & 31, wave = tid >> 5;
  const int r0 = blockIdx.x * 128;   // row base in T*B

  // async copy obs strip -> sA
  for (int i = tid; i < 128 * 32; i += 256) {
    int row = i >> 5, c4 = (i & 31) << 2;
    g2l_async_b128(&obs[(size_t)(r0 + row) * OO + c4], &sA[row * LDP + c4]);
  }
  wait_async0();
  __syncthreads();
  layer_pass(sA, w1, b1, sB, wave, lane, 0, 8, true);   // x1 = tanh(obs @ w1^T + b1)
  __syncthreads();
  layer_pass(sB, w2, b2, sA, wave, lane, 0, 8, true);   // x2 = tanh(x1 @ w2^T + b2)
  __syncthreads();
  for (int cc = 0; cc < 4; ++cc) {                      // xg chunk of 128 gate-cols
    layer_pass(sA, wih, bih, sB, wave, lane, cc * 128, 8, false);
    __syncthreads();
    for (int i = tid; i < 128 * 32; i += 256) {
      int row = i >> 5, c4 = (i & 31) << 2;
      l2g_async_b128(&xg[(size_t)(r0 + row) * G4 + cc * 128 + c4], &sB[row * LDP + c4]);
    }
    wait_async0();
    __syncthreads();
  }
}

// ---------------- Policy/value heads for one timestep (reads unmasked h_t) ----------------
__device__ __forceinline__ void heads_rows(
    const float* __restrict__ h, const float* __restrict__ act_t,
    const float* __restrict__ amw, const float* __restrict__ amb,
    const float* __restrict__ lstd, const float* __restrict__ crw,
    const float* __restrict__ crb,
    float* __restrict__ lp_t, float* __restrict__ ent_t, float* __restrict__ val_t,
    int r0, int tid)
{
  int row = tid >> 1, half = tid & 1;
  int grow = r0 + row;
  const float* hr = h + (size_t)grow * HH + half * 64;
  float pv = 0.f;
  float pm[AAD];
#pragma unroll
  for (int a = 0; a < AAD; ++a) pm[a] = 0.f;
  for (int c = 0; c < 64; ++c) {
    float hv = hr[c];
    int cc = half * 64 + c;
    pv += crw[cc] * hv;
#pragma unroll
    for (int a = 0; a < AAD; ++a) pm[a] += amw[a * HH + cc] * hv;
  }
  pv += __shfl_xor(pv, 1);
#pragma unroll
  for (int a = 0; a < AAD; ++a) pm[a] += __shfl_xor(pm[a], 1);
  if (half == 0) {
    float lp = 0.f, lj = 0.f, ent = 0.f;
#pragma unroll
    for (int a = 0; a < AAD; ++a) {
      float mean = pm[a] + amb[a];
      float ls = lstd[a];
      float act = act_t[(size_t)grow * AAD + a];
      float u = fminf(fmaxf(act, -1.f + 1e-6f), 1.f - 1e-6f);
      u = 0.5f * (log1pf(u) - log1pf(-u));                 // atanh (clipped)
      float z = (u - mean) * expf(-ls);
      lp += -0.5f * z * z - ls - 0.5f * LOG2PI_F;
      lj += logf(1.f - act * act + 1e-6f);
      ent += 0.5f + 0.5f * LOG2PI_F + ls;
    }
    lp_t[grow] = lp - lj;
    ent_t[grow] = ent + lj;
    val_t[grow] = pv + crb[0];
  }
}

__global__ void __launch_bounds__(256) heads_kernel(
    const float* __restrict__ h, const float* __restrict__ act_t,
    const float* __restrict__ amw, const float* __restrict__ amb,
    const float* __restrict__ lstd, const float* __restrict__ crw,
    const float* __restrict__ crb,
    float* __restrict__ lp_t, float* __restrict__ ent_t, float* __restrict__ val_t)
{
  heads_rows(h, act_t, amw, amb, lstd, crw, crb, lp_t, ent_t, val_t,
             blockIdx.x * 128, threadIdx.x);
}

// ---------------- One LSTM step. grid = (8 row-blocks, 4 gate-col blocks) ----------------
// Also computes heads for the PREVIOUS timestep (blockIdx.y==0, do_heads!=0) from h_in.
__global__ void __launch_bounds__(256) lstm_step_kernel(
    const float* __restrict__ xgt,     // [B,512] for this t (b_ih already folded in)
    const float* __restrict__ dones_t, // [B]
    const float* __restrict__ h_in,    // [B,H] unmasked h_{t-1} (or h0)
    const float* __restrict__ c_in,    // [B,H]
    float* __restrict__ h_out, float* __restrict__ c_out,
    const float* __restrict__ whh, const float* __restrict__ bhh,
    const float* __restrict__ act_prev,
    const float* __restrict__ amw, const float* __restrict__ amb,
    const float* __restrict__ lstd, const float* __restrict__ crw,
    const float* __restrict__ crb,
    float* __restrict__ lp_prev, float* __restrict__ ent_prev, float* __restrict__ val_prev,
    int do_heads)
{
  __shared__ float sH[128 * LDP];
  const int tid = threadIdx.x, lane = tid & 31, wave = tid >> 5;
  const int lh = lane & 15, hl = lane >> 4;
  const int r0 = blockIdx.x * 128;

  // async copy RAW h strip -> LDS; done-mask is folded into the A-operand reads below
  for (int i = tid; i < 128 * 32; i += 256) {
    int row = i >> 5, c4 = (i & 31) << 2;
    g2l_async_b128(&h_in[(size_t)(r0 + row) * HH + c4], &sH[row * LDP + c4]);
  }
  wait_async0();
  __syncthreads();

  // A-operand lane maps to row (wave*16 + lh) only -> one mask scalar per lane
  const float mlane = 1.f - dones_t[r0 + wave * 16 + lh];
  const float* Ab = sH + (wave * 16 + lh) * LDP + (hl << 1);
  for (int cti = 0; cti < 2; ++cti) {
    const int ct = blockIdx.y * 2 + cti;        // 0..7 -> 16 h-cols each
    const int colb = ct * 16 + lh;
    const float* Bi = whh + (size_t)(0   + colb) * HH + (hl << 1);
    const float* Bf = whh + (size_t)(128 + colb) * HH + (hl << 1);
    const float* Bg = whh + (size_t)(256 + colb) * HH + (hl << 1);
    const float* Bo = whh + (size_t)(384 + colb) * HH + (hl << 1);
    v8f ai = {}, af = {}, ag = {}, ao = {};
#pragma unroll 4
    for (int kb = 0; kb < 128; kb += 4) {
      v2f a = *(const v2f*)(Ab + kb);
      a *= mlane;                               // masked h rows
      ai = wmma4(a, *(const v2f*)(Bi + kb), ai);
      af = wmma4(a, *(const v2f*)(Bf + kb), af);
      ag = wmma4(a, *(const v2f*)(Bg + kb), ag);
      ao = wmma4(a, *(const v2f*)(Bo + kb), ao);
    }
    float bi_ = bhh[colb], bf_ = bhh[128 + colb], bg_ = bhh[256 + colb], bo_ = bhh[384 + colb];
#pragma unroll
    for (int v = 0; v < 8; ++v) {
      int grow = r0 + wave * 16 + v + 8 * hl;
      const float* xr = xgt + (size_t)grow * G4;
      float gi = ai[v] + xr[colb]       + bi_;
      float gf = af[v] + xr[128 + colb] + bf_;
      float gg = ag[v] + xr[256 + colb] + bg_;
      float go = ao[v] + xr[384 + colb] + bo_;
      float m  = 1.f - dones_t[grow];
      float cold = c_in[(size_t)grow * HH + colb] * m;
      float c2 = sigm(gf) * cold + sigm(gi) * fast_tanh(gg);
      float h2 = sigm(go) * fast_tanh(c2);
      c_out[(size_t)grow * HH + colb] = c2;
      h_out[(size_t)grow * HH + colb] = h2;
    }
  }

  // heads for previous timestep (uses unmasked h_in = h_{t-1}); uniform branch
  if (blockIdx.y == 0 && do_heads) {
    heads_rows(h_in, act_prev, amw, amb, lstd, crw, crb,
               lp_prev, ent_prev, val_prev, r0, tid);
  }
}

extern "C" void kernel_launch(void* const* d_in, const int* in_sizes, int n_in,
                              void* d_out, int out_size, void* d_ws, size_t ws_size,
                              hipStream_t stream) {
  (void)in_sizes; (void)n_in; (void)out_size; (void)ws_size;
  const float* obs  = (const float*)d_in[0];
  const float* act  = (const float*)d_in[1];
  const float* dn   = (const float*)d_in[2];
  const float* h0   = (const float*)d_in[3];
  const float* c0   = (const float*)d_in[4];
  const float* w1   = (const float*)d_in[5];
  const float* b1   = (const float*)d_in[6];
  const float* w2   = (const float*)d_in[7];
  const float* b2   = (const float*)d_in[8];
  const float* wih  = (const float*)d_in[9];
  const float* whh  = (const float*)d_in[10];
  const float* bih  = (const float*)d_in[11];
  const float* bhh  = (const float*)d_in[12];
  const float* amw  = (const float*)d_in[13];
  const float* amb  = (const float*)d_in[14];
  const float* lstd = (const float*)d_in[15];
  const float* crw  = (const float*)d_in[16];
  const float* crb  = (const float*)d_in[17];
  float* out = (float*)d_out;

  const size_t TB = (size_t)TT * BB;
  float* xg    = (float*)d_ws;                       // T*B x 512  (512 MiB)
  float* hbuf0 = xg + TB * (size_t)G4;               // B x H ping
  float* hbuf1 = hbuf0 + (size_t)BB * HH;            // B x H pong
  float* stc   = hbuf1 + (size_t)BB * HH;            // B x H cell state (in-place safe)

  enc_xg_kernel<<<(TT * BB) / 128, 256, 0, stream>>>(obs, w1, b1, w2, b2, wih, bih, xg);

  for (int t = 0; t < TT; ++t) {
    const float* hsrc = (t == 0) ? h0  : ((t & 1) ? hbuf1 : hbuf0);
    const float* csrc = (t == 0) ? c0  : stc;
    float* hdst = ((t + 1) & 1) ? hbuf1 : hbuf0;
    lstm_step_kernel<<<dim3(BB / 128, 4), 256, 0, stream>>>(
        xg + (size_t)t * BB * G4, dn + (size_t)t * BB,
        hsrc, csrc, hdst, stc,
        whh, bhh,
        act + (size_t)(t > 0 ? t - 1 : 0) * BB * AAD,
        amw, amb, lstd, crw, crb,
        out + (size_t)(t > 0 ? t - 1 : 0) * BB,
        out + TB + (size_t)(t > 0 ? t - 1 : 0) * BB,
        out + 2 * TB + (size_t)(t > 0 ? t - 1 : 0) * BB,
        t > 0 ? 1 : 0);
  }
  // heads for final step t = 255 (h_255 lives in hbuf0 since (255+1)&1 == 0)
  heads_kernel<<<BB / 128, 256, 0, stream>>>(
      hbuf0, act + (size_t)(TT - 1) * BB * AAD, amw, amb, lstd, crw, crb,
      out + (size_t)(TT - 1) * BB, out + TB + (size_t)(TT - 1) * BB,
      out + 2 * TB + (size_t)(TT - 1) * BB);

  (void)hipMemcpyAsync(out + 3 * TB, hbuf0, (size_t)BB * HH * sizeof(float),
                       hipMemcpyDeviceToDevice, stream);
  (void)hipMemcpyAsync(out + 3 * TB + (size_t)BB * HH, stc, (size_t)BB * HH * sizeof(float),
                       hipMemcpyDeviceToDevice, stream);
}